// GATv2FeatureExtractor_4947802325827
// MI455X (gfx1250) — compile-verified
//
#include <hip/hip_runtime.h>
#include <math.h>

typedef __attribute__((ext_vector_type(2))) float v2f;
typedef __attribute__((ext_vector_type(8))) float v8f;
typedef __attribute__((ext_vector_type(4))) unsigned int u32x4;
typedef __attribute__((ext_vector_type(8))) int i32x8;
typedef __attribute__((ext_vector_type(4))) int i32x4;

#define NEG_SLOPE_C 0.2f

// Monotonic fp32 atomic max (buffer must be initialized to -inf).
__device__ __forceinline__ void atomicMaxFloat(float* addr, float v) {
  if (v >= 0.0f) atomicMax((int*)addr, __float_as_int(v));
  else           atomicMin((unsigned int*)addr, __float_as_uint(v));
}

// ---------------------------------------------------------------------------
// TDM: DMA `nelem` fp32 from global `src` into LDS offset 0 (dynamic smem
// base). D# built per CDNA5 ISA ch.8: group0 = {count/lds_addr/global_addr/
// type=2}, group1 = {data_size=4B, 1-row tile of nelem elements}.
// This toolchain exposes the 6-arg builtin form (g0,g1,g2,g3,g4,cpol).
// Issue from one wave; caller must s_wait_tensorcnt + barrier.
// ---------------------------------------------------------------------------
__device__ __forceinline__ void tdm_load_row_to_lds0(const float* src, int nelem) {
  unsigned long long ga = (unsigned long long)(uintptr_t)src;
  u32x4 g0;
  g0.x = 1u;                                        // count=1, user mode
  g0.y = 0u;                                        // lds_addr = 0 (dyn smem base)
  g0.z = (unsigned int)(ga & 0xFFFFFFFFu);          // global_addr[31:0]
  g0.w = (unsigned int)((ga >> 32) & 0x01FFFFFFu)   // global_addr[56:32]
         | 0x80000000u;                             // type=2 ("image")
  i32x8 g1;
  g1[0] = (int)(2u << 16);                          // data_size=2 -> 4 bytes
  g1[1] = (int)(((unsigned)nelem & 0xFFFFu) << 16); // tensor_dim0[15:0]
  g1[2] = (int)(((unsigned)nelem >> 16) & 0xFFFFu)  // tensor_dim0[31:16]
          | (1 << 16);                              // tensor_dim1 = 1
  g1[3] = (int)(((unsigned)nelem & 0xFFFFu) << 16); // tile_dim0 = nelem
  g1[4] = 1;                                        // tile_dim1 = 1
  g1[5] = nelem;                                    // tensor_dim0_stride[31:0]
  g1[6] = 0;
  g1[7] = 0;
  i32x4 gz4 = {0, 0, 0, 0};
  i32x8 gz8 = {0, 0, 0, 0, 0, 0, 0, 0};
  __builtin_amdgcn_tensor_load_to_lds(g0, g1, gz4, gz4, gz8, 0);
}

// ---------------------------------------------------------------------------
// C[M,N] = A[M,K] @ W[K,N], fp32 row-major, V_WMMA_F32_16X16X4_F32.
// One wave per 16x16 output tile; 8 waves/block share the 16-row A tile in LDS.
// Requires: M%16==0, (N/16)%8==0, K%KCHUNK==0, KCHUNK%4==0, blockDim.x==256.
// ---------------------------------------------------------------------------
template <int KCHUNK>
__global__ __launch_bounds__(256) void gemm_wmma_f32(
    const float* __restrict__ A, const float* __restrict__ W,
    float* __restrict__ C, int M, int K, int N) {
  __shared__ __align__(16) float sA[16 * KCHUNK];
  const int tid  = threadIdx.x;
  const int lane = tid & 31;
  const int wid  = tid >> 5;
  const int ntiles    = N >> 4;
  const int colBlocks = ntiles >> 3;                 // 8 waves per block
  const int tm = blockIdx.x / colBlocks;
  const int tn = (blockIdx.x % colBlocks) * 8 + wid;
  const int l    = lane & 15;
  const int half = lane >> 4;

  v8f acc = {};
  for (int k0 = 0; k0 < K; k0 += KCHUNK) {
    __syncthreads();
    for (int i = tid; i < 16 * KCHUNK; i += 256) {
      int r = i / KCHUNK, c = i % KCHUNK;
      sA[i] = A[(size_t)(tm * 16 + r) * K + k0 + c];
    }
    if (k0 + KCHUNK < K)  // prefetch next A chunk (global_prefetch_b8)
      __builtin_prefetch(&A[(size_t)(tm * 16 + (tid & 15)) * K + k0 + KCHUNK +
                            (tid >> 4) * 4], 0, 0);
    __syncthreads();
    const float* Wp = W + (size_t)k0 * N + tn * 16 + l;
#pragma unroll 4
    for (int kk = 0; kk < KCHUNK; kk += 4) {
      v2f a = *(const v2f*)&sA[l * KCHUNK + kk + 2 * half];
      v2f b;
      b.x = Wp[(size_t)(kk + 2 * half + 0) * N];
      b.y = Wp[(size_t)(kk + 2 * half + 1) * N];
      acc = __builtin_amdgcn_wmma_f32_16x16x4_f32(
          /*neg_a=*/false, a, /*neg_b=*/false, b,
          /*c_mod=*/(short)0, acc, /*reuse_a=*/false, /*reuse_b=*/false);
    }
  }
  const int row0 = tm * 16 + half * 8;
  const int col  = tn * 16 + l;
#pragma unroll
  for (int r = 0; r < 8; r++) C[(size_t)(row0 + r) * N + col] = acc[r];
}

// ---------------------------------------------------------------------------
// Per-edge attention score. One wave per edge, 16 waves/block. We is staged
// in LDS by the Tensor Data Mover (wave 0); ee = edge_attr @ We computed on
// the fly to avoid materializing [E, H*C] in HBM. Dynamic smem layout:
//   [0, EDIM*HC)  : We   (TDM target, offset 0)
//   [EDIM*HC, +HC): att
// ---------------------------------------------------------------------------
template <int H, int C, int EDIM>
__global__ __launch_bounds__(512) void edge_score(
    const float* __restrict__ xl, const float* __restrict__ xr,
    const float* __restrict__ ea, const float* __restrict__ We,
    const float* __restrict__ att, const int* __restrict__ src,
    const int* __restrict__ dst, float* __restrict__ escore,
    float* __restrict__ mbuf, int E) {
  constexpr int HC = H * C;
  extern __shared__ float smem[];
  float* sWe  = smem;            // EDIM*HC floats at LDS offset 0
  float* sAtt = smem + EDIM * HC;

  const int wid = threadIdx.x >> 5;
  if (wid == 0) {                          // wave-uniform branch
    tdm_load_row_to_lds0(We, EDIM * HC);   // tensor_load_to_lds
    __builtin_amdgcn_s_wait_tensorcnt(0);
  }
  for (int i = threadIdx.x; i < HC; i += blockDim.x) sAtt[i] = att[i];
  __syncthreads();

  const int lane = threadIdx.x & 31;
  const int edge = blockIdx.x * (blockDim.x >> 5) + wid;
  if (edge >= E) return;
  const int s = src[edge], d = dst[edge];

  float aev[EDIM];
#pragma unroll
  for (int k = 0; k < EDIM; k++) aev[k] = ea[(size_t)edge * EDIM + k];

  float acc[H];
#pragma unroll
  for (int h = 0; h < H; h++) acc[h] = 0.f;

  const float* xls = xl + (size_t)s * HC;
  const float* xrd = xr + (size_t)d * HC;
#pragma unroll
  for (int j = 0; j < HC / 32; j++) {
    const int ch = j * 32 + lane;
    float eev = 0.f;
#pragma unroll
    for (int k = 0; k < EDIM; k++) eev = fmaf(aev[k], sWe[k * HC + ch], eev);
    float f = xls[ch] + xrd[ch] + eev;
    f = f > 0.f ? f : NEG_SLOPE_C * f;
    acc[(j * 32) / C] = fmaf(sAtt[ch], f, acc[(j * 32) / C]);  // head uniform per j
  }
#pragma unroll
  for (int h = 0; h < H; h++) {
    float v = acc[h];
    for (int off = 16; off > 0; off >>= 1) v += __shfl_xor(v, off, 32);
    if (lane == 0) {
      escore[(size_t)edge * H + h] = v;
      atomicMaxFloat(&mbuf[(size_t)d * H + h], v);
    }
  }
}

// exp(e - max) in place, atomic segment sum into sbuf (init to 0).
__global__ void edge_expsum(const int* __restrict__ dst, float* __restrict__ esc,
                            const float* __restrict__ m, float* __restrict__ s,
                            int E, int H) {
  int i = blockIdx.x * blockDim.x + threadIdx.x;
  if (i >= E * H) return;
  int e = i / H, h = i - e * H;
  int d = dst[e];
  float ex = expf(esc[i] - m[d * H + h]);
  esc[i] = ex;
  atomicAdd(&s[d * H + h], ex);
}

// out[dst] += alpha * xl[src] ; one wave per edge, f32 atomics (L2-resident).
template <int H, int C>
__global__ __launch_bounds__(512) void edge_aggregate(
    const float* __restrict__ xl, const float* __restrict__ ex,
    const float* __restrict__ ssum, const int* __restrict__ src,
    const int* __restrict__ dst, float* __restrict__ out, int E) {
  constexpr int HC = H * C;
  const int lane = threadIdx.x & 31;
  const int edge = blockIdx.x * (blockDim.x >> 5) + (threadIdx.x >> 5);
  if (edge >= E) return;
  const int s = src[edge], d = dst[edge];
  float alpha[H];
#pragma unroll
  for (int h = 0; h < H; h++)
    alpha[h] = ex[(size_t)edge * H + h] / ssum[(size_t)d * H + h];
  const float* xls = xl + (size_t)s * HC;
  float* od = out + (size_t)d * HC;
#pragma unroll
  for (int j = 0; j < HC / 32; j++) {
    const int ch = j * 32 + lane;
    atomicAdd(&od[ch], alpha[(j * 32) / C] * xls[ch]);
  }
}

__global__ void bias_relu(float* __restrict__ p, const float* __restrict__ b,
                          int n, int hc) {
  size_t i = (size_t)blockIdx.x * blockDim.x + threadIdx.x;
  if (i >= (size_t)n * hc) return;
  float v = p[i] + b[i % hc];
  p[i] = v > 0.f ? v : 0.f;
}

__global__ void fill_f32(float* __restrict__ p, float v, size_t n) {
  size_t i = (size_t)blockIdx.x * blockDim.x + threadIdx.x;
  if (i < n) p[i] = v;
}

// ---------------------------------------------------------------------------
extern "C" void kernel_launch(void* const* d_in, const int* in_sizes, int n_in,
                              void* d_out, int out_size, void* d_ws, size_t ws_size,
                              hipStream_t stream) {
  const int NN = 20000, E = 320000;
  const int HC1 = 512, HC2 = 256, H1 = 4;

  const float* x    = (const float*)d_in[0];
  const int*   ei   = (const int*)d_in[1];
  const float* ea   = (const float*)d_in[2];
  const float* Wl1  = (const float*)d_in[3];
  const float* Wr1  = (const float*)d_in[4];
  const float* We1  = (const float*)d_in[5];
  const float* att1 = (const float*)d_in[6];
  const float* b1   = (const float*)d_in[7];
  const float* Wl2  = (const float*)d_in[8];
  const float* Wr2  = (const float*)d_in[9];
  const float* We2  = (const float*)d_in[10];
  const float* att2 = (const float*)d_in[11];
  const float* b2   = (const float*)d_in[12];
  float* out = (float*)d_out;
  float* ws  = (float*)d_ws;
  const int* src = ei;
  const int* dst = ei + E;

  size_t o = 0;
  float* xl1  = ws + o; o += (size_t)NN * HC1;
  float* xr1  = ws + o; o += (size_t)NN * HC1;
  float* hbuf = ws + o; o += (size_t)NN * HC1;
  float* xl2  = ws + o; o += (size_t)NN * HC2;
  float* xr2  = ws + o; o += (size_t)NN * HC2;
  float* ebuf = ws + o; o += (size_t)E * H1;
  float* mbuf = ws + o; o += (size_t)NN * H1;
  float* sbuf = ws + o; o += (size_t)NN * H1;
  (void)o; (void)ws_size; (void)in_sizes; (void)n_in; (void)out_size;

  const float NEGINF = -INFINITY;
  const int EW = E / 16;  // edge blocks: 16 waves of 32 per block

  // ---------------- Layer 1 : H=4, C=128 ----------------
  {
    int grid = (NN / 16) * ((HC1 / 16) / 8);  // 5000
    gemm_wmma_f32<64><<<grid, 256, 0, stream>>>(x, Wl1, xl1, NN, 64, HC1);
    gemm_wmma_f32<64><<<grid, 256, 0, stream>>>(x, Wr1, xr1, NN, 64, HC1);
  }
  fill_f32<<<(NN * H1 + 255) / 256, 256, 0, stream>>>(mbuf, NEGINF, (size_t)NN * H1);
  fill_f32<<<(NN * H1 + 255) / 256, 256, 0, stream>>>(sbuf, 0.f, (size_t)NN * H1);
  fill_f32<<<(int)(((size_t)NN * HC1 + 255) / 256), 256, 0, stream>>>(hbuf, 0.f, (size_t)NN * HC1);
  edge_score<4, 128, 16><<<EW, 512, (16 * HC1 + HC1) * sizeof(float), stream>>>(
      xl1, xr1, ea, We1, att1, src, dst, ebuf, mbuf, E);
  edge_expsum<<<(E * H1 + 255) / 256, 256, 0, stream>>>(dst, ebuf, mbuf, sbuf, E, H1);
  edge_aggregate<4, 128><<<EW, 512, 0, stream>>>(xl1, ebuf, sbuf, src, dst, hbuf, E);
  bias_relu<<<(int)(((size_t)NN * HC1 + 255) / 256), 256, 0, stream>>>(hbuf, b1, NN, HC1);

  // ---------------- Layer 2 : H=1, C=256 ----------------
  {
    int grid = (NN / 16) * ((HC2 / 16) / 8);  // 2500
    gemm_wmma_f32<64><<<grid, 256, 0, stream>>>(hbuf, Wl2, xl2, NN, HC1, HC2);
    gemm_wmma_f32<64><<<grid, 256, 0, stream>>>(hbuf, Wr2, xr2, NN, HC1, HC2);
  }
  fill_f32<<<(NN + 255) / 256, 256, 0, stream>>>(mbuf, NEGINF, (size_t)NN);
  fill_f32<<<(NN + 255) / 256, 256, 0, stream>>>(sbuf, 0.f, (size_t)NN);
  fill_f32<<<(int)(((size_t)NN * HC2 + 255) / 256), 256, 0, stream>>>(out, 0.f, (size_t)NN * HC2);
  edge_score<1, 256, 16><<<EW, 512, (16 * HC2 + HC2) * sizeof(float), stream>>>(
      xl2, xr2, ea, We2, att2, src, dst, ebuf, mbuf, E);
  edge_expsum<<<(E + 255) / 256, 256, 0, stream>>>(dst, ebuf, mbuf, sbuf, E, 1);
  edge_aggregate<1, 256><<<EW, 512, 0, stream>>>(xl2, ebuf, sbuf, src, dst, out, E);
  bias_relu<<<(int)(((size_t)NN * HC2 + 255) / 256), 256, 0, stream>>>(out, b2, NN, HC2);
}